// WaveNet_9826885173593
// MI455X (gfx1250) — compile-verified
//
#include <hip/hip_runtime.h>
#include <math.h>

#define B_    4
#define T_    8192
#define TT    64      // time tile per block
#define KP    160     // dwords per k-pair row in LDS (conflict-free for b64)
#define NLAY  20

typedef float v2f __attribute__((ext_vector_type(2)));
typedef float v8f __attribute__((ext_vector_type(8)));

// D = A(16x4) * B(4x16) + C, fp32 WMMA
__device__ __forceinline__ v8f wmma4(v2f a, v2f b, v8f c) {
  return __builtin_amdgcn_wmma_f32_16x16x4_f32(false, a, false, b, (short)0, c,
                                               false, false);
}

// LDS index for k-pair-interleaved activation tile: (k,t) -> (k/2)*KP + 2t + (k&1)
__device__ __forceinline__ int lidx(int k, int t) {
  return ((k >> 1) * KP) + (t << 1) + (k & 1);
}

// A-operand (16x4 fp32): row-major [M][K] weights, one aligned b64 per lane
__device__ __forceinline__ v2f gA(const float* __restrict__ W, int rowStride,
                                  int m0, int k0) {
  int lane = threadIdx.x & 31;
  int row  = m0 + (lane & 15);
  int kb   = k0 + ((lane >> 4) << 1);          // even
  return *(const v2f*)(W + row * rowStride + kb);
}

// B-operand (4x16 fp32) from k-pair-interleaved LDS tile: one aligned ds b64
__device__ __forceinline__ v2f ldsB(const float* __restrict__ base, int k0,
                                    int colBase) {
  int lane = threadIdx.x & 31;
  int c    = colBase + (lane & 15);
  int kp   = (k0 >> 1) + (lane >> 4);          // k-pair row
  return *(const v2f*)(base + kp * KP + (c << 1));
}

// ------- weight prep: split tap-interleaved conv/embed weights ---------------
__global__ __launch_bounds__(256) void prep_kernel(
    const float* __restrict__ cw, const float* __restrict__ ew,
    float* __restrict__ w0, float* __restrict__ w1,
    float* __restrict__ e0, float* __restrict__ e1) {
  int i = blockIdx.x * 256 + threadIdx.x;      // pair index
  if (i < NLAY * 128 * 64) {
    w0[i] = cw[2 * i];
    w1[i] = cw[2 * i + 1];
  }
  if (i < 64 * 256) {
    e0[i] = ew[2 * i];
    e1[i] = ew[2 * i + 1];
  }
}

// ------- embed: x = W0@in[t-1] + W1@in[t] + b  (256 -> 64 ch) ----------------
__global__ __launch_bounds__(256) void embed_kernel(
    const float* __restrict__ input, const float* __restrict__ e0,
    const float* __restrict__ e1, const float* __restrict__ eb,
    float* __restrict__ xout) {
  extern __shared__ float sm[];  // 128 k-pairs x KP; logical cols 0..64 = t0-1..t0+63
  const int tid = threadIdx.x;
  const int b   = blockIdx.y;
  const int t0  = blockIdx.x * TT;
  const int lane = tid & 31, wv = tid >> 5;
  const int hi = lane >> 4, lo = lane & 15;

  for (int i = tid; i < 256 * (TT + 1); i += 256) {
    int c = i / (TT + 1), lc = i % (TT + 1);
    int t = t0 - 1 + lc;
    sm[lidx(c, lc)] = (t >= 0) ? input[(b * 256 + c) * T_ + t] : 0.f;
  }
  __syncthreads();

  for (int j = 0; j < 2; ++j) {
    int tile = wv + 8 * j;          // 16 tiles: 4 M x 4 N
    int m0 = (tile >> 2) * 16;
    int n0 = (tile & 3) * 16;
    v8f acc;
#pragma unroll
    for (int r = 0; r < 8; ++r) acc[r] = eb[m0 + r + hi * 8];
    for (int k0 = 0; k0 < 256; k0 += 4) {
      v2f a0 = gA(e0, 256, m0, k0);                 // tap0 -> t-1 (lds col n)
      v2f a1 = gA(e1, 256, m0, k0);                 // tap1 -> t   (lds col n+1)
      acc = wmma4(a0, ldsB(sm, k0, n0), acc);
      acc = wmma4(a1, ldsB(sm, k0, n0 + 1), acc);
    }
#pragma unroll
    for (int r = 0; r < 8; ++r)
      xout[(b * 64 + m0 + r + hi * 8) * T_ + t0 + n0 + lo] = acc[r];
  }
}

// ------- one WaveNet layer ---------------------------------------------------
__global__ __launch_bounds__(256) void layer_kernel(
    const float* __restrict__ xin, float* __restrict__ xout,
    const float* __restrict__ cond, float* __restrict__ skp,
    const float* __restrict__ w0, const float* __restrict__ w1,  // [128][64] taps
    const float* __restrict__ cb,                                // conv bias [128]
    const float* __restrict__ dw, const float* __restrict__ db,  // cond [128][128]
    const float* __restrict__ rw, const float* __restrict__ rb,  // res  [64][64]
    const float* __restrict__ sw, const float* __restrict__ sb,  // skip [128][64]
    int dil, int first) {
  extern __shared__ float sm[];
  float* s_xc = sm;                 // 32 kp x KP : x[t]
  float* s_xd = s_xc + 32 * KP;     // 32 kp x KP : x[t-d]
  float* s_cd = s_xd + 32 * KP;     // 64 kp x KP : cond
  float* s_h  = s_cd + 64 * KP;     // 64 kp x KP : h, then z in channels 0..63

  const int tid = threadIdx.x;
  const int b   = blockIdx.y;
  const int t0  = blockIdx.x * TT;
  const int lane = tid & 31, wv = tid >> 5;
  const int hi = lane >> 4, lo = lane & 15;

  for (int i = tid; i < 64 * TT; i += 256) {
    int c = i >> 6, t = i & (TT - 1);
    s_xc[lidx(c, t)] = xin[(b * 64 + c) * T_ + t0 + t];
    int td = t0 - dil + t;
    s_xd[lidx(c, t)] = (td >= 0) ? xin[(b * 64 + c) * T_ + td] : 0.f;
  }
  for (int i = tid; i < 128 * TT; i += 256) {
    int c = i >> 6, t = i & (TT - 1);
    s_cd[lidx(c, t)] = cond[(b * 128 + c) * T_ + t0 + t];
  }
  __syncthreads();

  // h[128x64] = W0@x(t-d) + W1@x(t) + Wc@cond + biases; wave wv owns 16 rows
  {
    int m0 = wv * 16;
    v8f acc[4];
#pragma unroll
    for (int ni = 0; ni < 4; ++ni)
#pragma unroll
      for (int r = 0; r < 8; ++r) {
        int m = m0 + r + hi * 8;
        acc[ni][r] = cb[m] + db[m];
      }
    for (int k0 = 0; k0 < 64; k0 += 4) {
      v2f a0 = gA(w0, 64, m0, k0);
      v2f a1 = gA(w1, 64, m0, k0);
#pragma unroll
      for (int ni = 0; ni < 4; ++ni) {
        acc[ni] = wmma4(a0, ldsB(s_xd, k0, ni * 16), acc[ni]);
        acc[ni] = wmma4(a1, ldsB(s_xc, k0, ni * 16), acc[ni]);
      }
    }
    for (int k0 = 0; k0 < 128; k0 += 4) {
      v2f a = gA(dw, 128, m0, k0);
#pragma unroll
      for (int ni = 0; ni < 4; ++ni)
        acc[ni] = wmma4(a, ldsB(s_cd, k0, ni * 16), acc[ni]);
    }
#pragma unroll
    for (int ni = 0; ni < 4; ++ni)
#pragma unroll
      for (int r = 0; r < 8; ++r) {
        int m = m0 + r + hi * 8;
        s_h[lidx(m, ni * 16 + lo)] = acc[ni][r];
      }
  }
  __syncthreads();

  // gated activation: z = tanh(h[0:64]) * sigmoid(h[64:128]), in place ch 0..63
  for (int i = tid; i < 64 * TT; i += 256) {
    int c = i >> 6, t = i & (TT - 1);
    float tz = s_h[lidx(c, t)];
    float sz = s_h[lidx(c + 64, t)];
    s_h[lidx(c, t)] = tanhf(tz) * (1.f / (1.f + __expf(-sz)));
  }
  __syncthreads();

  // res (16 tiles) + skip (32 tiles) round-robined over 8 waves
  for (int j = 0; j < 6; ++j) {
    int tile = wv + 8 * j;
    int isRes = (tile < 16);
    int tt = isRes ? tile : tile - 16;
    int m0 = (tt >> 2) * 16;
    int n0 = (tt & 3) * 16;
    v8f acc;
#pragma unroll
    for (int r = 0; r < 8; ++r) {
      int m = m0 + r + hi * 8;
      int n = n0 + lo;
      if (isRes)
        acc[r] = rb[m] + s_xc[lidx(m, n)];                        // residual
      else
        acc[r] = sb[m] + (first ? 0.f : skp[(b * 128 + m) * T_ + t0 + n]);
    }
    const float* W = isRes ? rw : sw;
    for (int k0 = 0; k0 < 64; k0 += 4)
      acc = wmma4(gA(W, 64, m0, k0), ldsB(s_h, k0, n0), acc);
#pragma unroll
    for (int r = 0; r < 8; ++r) {
      int m = m0 + r + hi * 8;
      int n = t0 + n0 + lo;
      if (isRes) xout[(b * 64 + m) * T_ + n] = acc[r];
      else       skp[(b * 128 + m) * T_ + n] = acc[r];
    }
  }
}

// ------- head: out = P2 @ relu(P1 @ relu(skip) + b1) + b2 --------------------
__global__ __launch_bounds__(256) void head_kernel(
    const float* __restrict__ skp, const float* __restrict__ w1,
    const float* __restrict__ b1, const float* __restrict__ w2,
    const float* __restrict__ b2, float* __restrict__ out) {
  extern __shared__ float sm[];
  float* s_s = sm;                 // 64 kp x KP : relu(skip)
  float* s_z = sm + 64 * KP;       // 64 kp x KP : relu(P1@s+b1)

  const int tid = threadIdx.x;
  const int b   = blockIdx.y;
  const int t0  = blockIdx.x * TT;
  const int lane = tid & 31, wv = tid >> 5;
  const int hi = lane >> 4, lo = lane & 15;

  for (int i = tid; i < 128 * TT; i += 256) {
    int c = i >> 6, t = i & (TT - 1);
    s_s[lidx(c, t)] = fmaxf(skp[(b * 128 + c) * T_ + t0 + t], 0.f);
  }
  __syncthreads();

  {  // z1: wave wv owns 16 rows x 64 cols
    int m0 = wv * 16;
    v8f acc[4];
#pragma unroll
    for (int ni = 0; ni < 4; ++ni)
#pragma unroll
      for (int r = 0; r < 8; ++r) acc[ni][r] = b1[m0 + r + hi * 8];
    for (int k0 = 0; k0 < 128; k0 += 4) {
      v2f a = gA(w1, 128, m0, k0);
#pragma unroll
      for (int ni = 0; ni < 4; ++ni)
        acc[ni] = wmma4(a, ldsB(s_s, k0, ni * 16), acc[ni]);
    }
#pragma unroll
    for (int ni = 0; ni < 4; ++ni)
#pragma unroll
      for (int r = 0; r < 8; ++r) {
        int m = m0 + r + hi * 8;
        s_z[lidx(m, ni * 16 + lo)] = fmaxf(acc[ni][r], 0.f);
      }
  }
  __syncthreads();

  for (int jj = 0; jj < 2; ++jj) {  // 256 output rows: wave does rows wv, wv+8
    int m0 = (wv + 8 * jj) * 16;
    v8f acc[4];
#pragma unroll
    for (int ni = 0; ni < 4; ++ni)
#pragma unroll
      for (int r = 0; r < 8; ++r) acc[ni][r] = b2[m0 + r + hi * 8];
    for (int k0 = 0; k0 < 128; k0 += 4) {
      v2f a = gA(w2, 128, m0, k0);
#pragma unroll
      for (int ni = 0; ni < 4; ++ni)
        acc[ni] = wmma4(a, ldsB(s_z, k0, ni * 16), acc[ni]);
    }
#pragma unroll
    for (int ni = 0; ni < 4; ++ni)
#pragma unroll
      for (int r = 0; r < 8; ++r)
        out[(b * 256 + m0 + r + hi * 8) * T_ + t0 + ni * 16 + lo] = acc[ni][r];
  }
}

extern "C" void kernel_launch(void* const* d_in, const int* in_sizes, int n_in,
                              void* d_out, int out_size, void* d_ws, size_t ws_size,
                              hipStream_t stream) {
  const float* input   = (const float*)d_in[0];
  const float* cond    = (const float*)d_in[1];
  const float* embed_w = (const float*)d_in[2];
  const float* embed_b = (const float*)d_in[3];
  const float* conv_w  = (const float*)d_in[4];
  const float* conv_b  = (const float*)d_in[5];
  const float* cond_w  = (const float*)d_in[6];
  const float* cond_b  = (const float*)d_in[7];
  const float* res_w   = (const float*)d_in[8];
  const float* res_b   = (const float*)d_in[9];
  const float* skip_w  = (const float*)d_in[10];
  const float* skip_b  = (const float*)d_in[11];
  const float* p1_w    = (const float*)d_in[12];
  const float* p1_b    = (const float*)d_in[13];
  const float* p2_w    = (const float*)d_in[14];
  const float* p2_b    = (const float*)d_in[15];
  float* out = (float*)d_out;

  // workspace: x ping-pong (2x8MB) + skip (16MB) + split weights (~1.4MB)
  size_t nx = (size_t)B_ * 64 * T_;
  float* x0   = (float*)d_ws;
  float* x1   = x0 + nx;
  float* skip = x1 + nx;
  float* w0   = skip + (size_t)B_ * 128 * T_;
  float* w1   = w0 + (size_t)NLAY * 128 * 64;
  float* e0   = w1 + (size_t)NLAY * 128 * 64;
  float* e1   = e0 + (size_t)64 * 256;

  dim3 grid(T_ / TT, B_), block(256);

  prep_kernel<<<dim3((NLAY * 128 * 64 + 255) / 256), block, 0, stream>>>(
      conv_w, embed_w, w0, w1, e0, e1);

  embed_kernel<<<grid, block, 128 * KP * sizeof(float), stream>>>(
      input, e0, e1, embed_b, x0);

  const float* xin = x0;
  float* xout = x1;
  for (int i = 0; i < NLAY; ++i) {
    int dil = 1 << (i % 10);
    layer_kernel<<<grid, block, 192 * KP * sizeof(float), stream>>>(
        xin, xout, cond, skip,
        w0 + (size_t)i * 128 * 64, w1 + (size_t)i * 128 * 64,
        conv_b + (size_t)i * 128,
        cond_w + (size_t)i * 128 * 128, cond_b + (size_t)i * 128,
        res_w + (size_t)i * 64 * 64,    res_b + (size_t)i * 64,
        skip_w + (size_t)i * 128 * 64,  skip_b + (size_t)i * 128,
        dil, (i == 0) ? 1 : 0);
    const float* tmp = xin;
    xin  = xout;
    xout = (float*)tmp;
  }

  head_kernel<<<grid, block, 128 * KP * sizeof(float), stream>>>(
      skip, p1_w, p1_b, p2_w, p2_b, out);
}